// SelfAttention2d_44289702756397
// MI455X (gfx1250) — compile-verified
//
#include <hip/hip_runtime.h>
#include <math.h>

// MI455X / gfx1250: wave32, WMMA 16x16, async global->LDS staging.
// Fused SAGAN self-attention: s = F^T G (16x1024x4096 per batch) is never
// materialized in HBM; flash-style two-pass softmax with recompute.

typedef float v2f __attribute__((ext_vector_type(2)));
typedef float v4f __attribute__((ext_vector_type(4)));
typedef float v8f __attribute__((ext_vector_type(8)));
typedef int   v4i __attribute__((ext_vector_type(4)));

#define AS1 __attribute__((address_space(1)))
#define AS3 __attribute__((address_space(3)))

// D = A(16x4 f32) * B(4x16 f32) + C(16x16 f32)
#define WMMA_F32X4(a, b, c) \
    __builtin_amdgcn_wmma_f32_16x16x4_f32(false, (a), false, (b), (short)0, (c), false, false)

#ifndef __has_builtin
#define __has_builtin(x) 0
#endif

#if __has_builtin(__builtin_amdgcn_global_load_async_to_lds_b128)
#define USE_ASYNC_LDS 1
#else
#define USE_ASYNC_LDS 0
#endif

// Copy 16 bytes global -> LDS. Async (ASYNCcnt-tracked DMA) when available.
// AS1 pointer value == flat global address; AS3 pointer value == low 32 bits
// of the flat LDS address (the LDS byte offset), hence the integer casts.
__device__ __forceinline__ void copy16_to_lds(const float* g, float* l)
{
#if USE_ASYNC_LDS
    __builtin_amdgcn_global_load_async_to_lds_b128(
        (AS1 v4i*)(unsigned long long)g,
        (AS3 v4i*)(unsigned long long)l, 0, 0);
#else
    *(v4f*)l = *(const v4f*)g;
#endif
}

#if USE_ASYNC_LDS
#if __has_builtin(__builtin_amdgcn_s_wait_asynccnt)
#define WAIT_ASYNC(n) __builtin_amdgcn_s_wait_asynccnt(n)
#else
#define WAIT_ASYNC(n) asm volatile("s_wait_asynccnt %0" ::"i"(n) : "memory")
#endif
#else
#define WAIT_ASYNC(n)
#endif

// ---------------------------------------------------------------------------
// G[b][k][n] = sum_c x[b][c][n]*Wg[k][c] + bg[k]   (k<16, n<4096), float4 lanes
// ---------------------------------------------------------------------------
__global__ void proj_g_kernel(const float* __restrict__ x, const float* __restrict__ W,
                              const float* __restrict__ bias, float* __restrict__ G)
{
    int n = 4 * (blockIdx.x * 256 + threadIdx.x);  // grid.x=4 -> n in [0,4096)
    int k = blockIdx.y;                             // uniform -> scalar W loads
    int b = blockIdx.z;
    const float* xp = x + (size_t)b * 128 * 4096 + n;
    const float* wp = W + k * 128;
    v4f acc = {};
#pragma unroll 8
    for (int c = 0; c < 128; ++c)
        acc += (*(const v4f*)(xp + (size_t)c * 4096)) * wp[c];
    *(v4f*)(G + ((size_t)b * 16 + k) * 4096 + n) = acc + bias[k];
}

// ---------------------------------------------------------------------------
// Conv + 2x2 maxpool: out[b][k][m] = max_{2x2}(conv) + bias, 4 m per thread.
// ---------------------------------------------------------------------------
__global__ void proj_pool_kernel(const float* __restrict__ x, const float* __restrict__ W,
                                 const float* __restrict__ bias, float* __restrict__ out,
                                 int OC)
{
    int m0 = 4 * threadIdx.x;                  // grid.x=1, 256 thr * 4 = 1024 m
    int k  = blockIdx.y;
    int b  = blockIdx.z;
    int hm = m0 >> 5, wm = m0 & 31;
    int p0 = hm * 128 + wm * 2;                // 8 consecutive cols, 2 rows
    const float* xp = x + (size_t)b * 128 * 4096 + p0;
    const float* wp = W + k * 128;
    v4f aA = {}, aB = {}, aC = {}, aD = {};    // row0 cols0-3/4-7, row1 cols0-3/4-7
#pragma unroll 4
    for (int c = 0; c < 128; ++c) {
        const float* xc = xp + (size_t)c * 4096;
        float w = wp[c];
        aA += (*(const v4f*)(xc +  0)) * w;
        aB += (*(const v4f*)(xc +  4)) * w;
        aC += (*(const v4f*)(xc + 64)) * w;
        aD += (*(const v4f*)(xc + 68)) * w;
    }
    v4f res;
    res.x = fmaxf(fmaxf(aA.x, aA.y), fmaxf(aC.x, aC.y));
    res.y = fmaxf(fmaxf(aA.z, aA.w), fmaxf(aC.z, aC.w));
    res.z = fmaxf(fmaxf(aB.x, aB.y), fmaxf(aD.x, aD.y));
    res.w = fmaxf(fmaxf(aB.z, aB.w), fmaxf(aD.z, aD.w));
    *(v4f*)(out + ((size_t)b * OC + k) * 1024 + m0) = res + bias[k];
}

// ---------------------------------------------------------------------------
// Flash pass 1: rowMax/rowInvZ of s = F^T G. 8 waves stream 256 n-tiles,
// s-tiles via 4x V_WMMA_F32_16X16X4_F32, online (max,sumexp) per C-frag row.
// ---------------------------------------------------------------------------
__global__ void row_stats_kernel(const float* __restrict__ F, const float* __restrict__ G,
                                 float* __restrict__ rowMax, float* __restrict__ rowInvZ)
{
    int mTile = blockIdx.x;
    int b     = blockIdx.y;
    int tid   = threadIdx.x;
    int wave  = tid >> 5;
    int lane  = tid & 31;
    int half  = lane >> 4;
    int l16   = lane & 15;

    const float* Fb = F + (size_t)b * 16 * 1024;
    const float* Gb = G + (size_t)b * 16 * 4096;
    int mG = mTile * 16 + l16;

    v2f afrag[4];                               // A[M=m][K=k] = F[k][m]
#pragma unroll
    for (int kk = 0; kk < 4; ++kk) {
        int k0 = kk * 4 + 2 * half;
        afrag[kk].x = Fb[(k0 + 0) * 1024 + mG];
        afrag[kk].y = Fb[(k0 + 1) * 1024 + mG];
    }

    float runMax[8], runSum[8];
#pragma unroll
    for (int r = 0; r < 8; ++r) { runMax[r] = -INFINITY; runSum[r] = 0.f; }

    for (int nt = wave; nt < 256; nt += 8) {
        int nG = nt * 16 + l16;
        v8f c = {};
#pragma unroll
        for (int kk = 0; kk < 4; ++kk) {
            int k0 = kk * 4 + 2 * half;
            v2f bfr;
            bfr.x = Gb[(k0 + 0) * 4096 + nG];
            bfr.y = Gb[(k0 + 1) * 4096 + nG];
            c = WMMA_F32X4(afrag[kk], bfr, c);
        }
#pragma unroll
        for (int r = 0; r < 8; ++r) {           // row M=r+8*half, col N=l16
            float v  = c[r];
            float nm = fmaxf(runMax[r], v);
            runSum[r] = runSum[r] * __expf(runMax[r] - nm) + __expf(v - nm);
            runMax[r] = nm;
        }
    }

    __shared__ float redM[16][129];
    __shared__ float redS[16][129];
    int col = wave * 16 + l16;
#pragma unroll
    for (int r = 0; r < 8; ++r) {
        int row = r + 8 * half;
        redM[row][col] = runMax[r];
        redS[row][col] = runSum[r];
    }
    __syncthreads();
    if (tid < 16) {
        float M = -INFINITY, S = 0.f;
        for (int c2 = 0; c2 < 128; ++c2) {
            float m2 = redM[tid][c2], s2 = redS[tid][c2];
            float nm = fmaxf(M, m2);
            S = S * __expf(M - nm) + s2 * __expf(m2 - nm);
            M = nm;
        }
        int m = mTile * 16 + tid;
        rowMax [(size_t)b * 1024 + m] = M;
        rowInvZ[(size_t)b * 1024 + m] = 1.f / S;
    }
}

// ---------------------------------------------------------------------------
// In-place: V[b][r][m] *= rowInvZ[b][m]  (V becomes U, r-major)
// ---------------------------------------------------------------------------
__global__ void scale_v_kernel(float* __restrict__ V, const float* __restrict__ rowInvZ)
{
    int m = blockIdx.x * 256 + threadIdx.x;    // grid.x=4
    int r = blockIdx.y;
    int b = blockIdx.z;
    size_t i = ((size_t)b * 64 + r) * 1024 + m;
    V[i] *= rowInvZ[(size_t)b * 1024 + m];
}

// ---------------------------------------------------------------------------
// Flash pass 2: o[b][r][n] = sum_m U[r][m] * exp(s[m][n] - rowMax[m]).
// Per block (batch, 128 n-cols): U m-tiles (64x16 = 4KB) double-buffered in
// LDS via async global->LDS B128 DMA; P staged transposed so B-frags read b64.
// ---------------------------------------------------------------------------
__global__ void attend_kernel(const float* __restrict__ F, const float* __restrict__ G,
                              const float* __restrict__ U, const float* __restrict__ rowMax,
                              float* __restrict__ O)
{
    int nBlk = blockIdx.x;
    int b    = blockIdx.y;
    int tid  = threadIdx.x;
    int wave = tid >> 5;
    int lane = tid & 31;
    int half = lane >> 4;
    int l16  = lane & 15;
    int nG   = nBlk * 128 + wave * 16 + l16;

    const float* Fb = F      + (size_t)b * 16 * 1024;
    const float* Gb = G      + (size_t)b * 16 * 4096;
    const float* Ub = U      + (size_t)b * 64 * 1024;   // r-major [64][1024]
    const float* Mb = rowMax + (size_t)b * 1024;

    // U tile buffers: 64 r-rows x 16 m, rows padded to 20 floats (80B keeps
    // 16B DMA alignment; 20*r mod 64 hits distinct banks). P tile transposed
    // [n][m], rows padded to 18 floats.
    __shared__ float Ust[2][64][20];
    __shared__ float pbufT[8][16][18];

    int srow = tid >> 2;                 // staging: thread -> (r-row, 16B chunk)
    int sch  = (tid & 3) * 4;

    copy16_to_lds(Ub + (size_t)srow * 1024 + sch, &Ust[0][srow][sch]);

    v2f gfrag[4];                        // B-frag of s-tile, m-loop invariant
#pragma unroll
    for (int kk = 0; kk < 4; ++kk) {
        int k0 = kk * 4 + 2 * half;
        gfrag[kk].x = Gb[(k0 + 0) * 4096 + nG];
        gfrag[kk].y = Gb[(k0 + 1) * 4096 + nG];
    }

    v8f zero = {};
    v8f oacc[4];
#pragma unroll
    for (int t = 0; t < 4; ++t) oacc[t] = zero;

    for (int mTile = 0; mTile < 64; ++mTile) {
        int mBase = mTile * 16;
        int cur   = mTile & 1;

        if (mTile < 63)                  // prefetch next U tile into other buffer
            copy16_to_lds(Ub + (size_t)srow * 1024 + mBase + 16 + sch,
                          &Ust[cur ^ 1][srow][sch]);

        // s tile = F^T G (16m x 16n), K=16 via 4 K4 WMMAs
        v8f c = {};
#pragma unroll
        for (int kk = 0; kk < 4; ++kk) {
            int k0 = kk * 4 + 2 * half;
            v2f af;
            af.x = Fb[(k0 + 0) * 1024 + mBase + l16];
            af.y = Fb[(k0 + 1) * 1024 + mBase + l16];
            c = WMMA_F32X4(af, gfrag[kk], c);
        }

        // P = exp(s - rowMax), stored transposed [n][m] as b64 pairs
        float pv[8];
#pragma unroll
        for (int r = 0; r < 8; ++r)
            pv[r] = __expf(c[r] - Mb[mBase + r + 8 * half]);
#pragma unroll
        for (int j = 0; j < 4; ++j) {
            v2f t; t.x = pv[2 * j]; t.y = pv[2 * j + 1];
            *(v2f*)&pbufT[wave][l16][8 * half + 2 * j] = t;
        }

        // current U tile complete (async ops retire in order: <=1 leaves it done)
        if (mTile < 63) { WAIT_ASYNC(1); } else { WAIT_ASYNC(0); }
        __syncthreads();                 // U tile + P visible to all waves

        // o(64 x 16n) += U(64 x 16m) * P(16m x 16n)
#pragma unroll
        for (int kk = 0; kk < 4; ++kk) {
            int kc = kk * 4 + 2 * half;
            v2f bf2 = *(const v2f*)&pbufT[wave][l16][kc];
#pragma unroll
            for (int rt = 0; rt < 4; ++rt) {
                v2f uf = *(const v2f*)&Ust[cur][rt * 16 + l16][kc];
                oacc[rt] = WMMA_F32X4(uf, bf2, oacc[rt]);
            }
        }
        __syncthreads();                 // buffers free before next overwrite
    }

#pragma unroll
    for (int rt = 0; rt < 4; ++rt)
#pragma unroll
        for (int vr = 0; vr < 8; ++vr) {
            int r = rt * 16 + vr + 8 * half;
            O[((size_t)b * 64 + r) * 4096 + nG] = oacc[rt][vr];
        }
}

// ---------------------------------------------------------------------------
// out = gamma*(Wo @ o + bo) + x, float4 lanes
// ---------------------------------------------------------------------------
__global__ void out_proj_kernel(const float* __restrict__ O, const float* __restrict__ Wo,
                                const float* __restrict__ bo, const float* __restrict__ gamma,
                                const float* __restrict__ x, float* __restrict__ out)
{
    int n   = 4 * (blockIdx.x * 256 + threadIdx.x);  // grid.x=4
    int cch = blockIdx.y;
    int b   = blockIdx.z;
    const float* Ob = O + (size_t)b * 64 * 4096 + n;
    const float* wp = Wo + cch * 64;
    v4f acc = {};
#pragma unroll 8
    for (int r = 0; r < 64; ++r)
        acc += (*(const v4f*)(Ob + (size_t)r * 4096)) * wp[r];
    size_t idx = ((size_t)b * 128 + cch) * 4096 + n;
    v4f xv = *(const v4f*)(x + idx);
    *(v4f*)(out + idx) = (acc + bo[cch]) * gamma[0] + xv;
}

// ---------------------------------------------------------------------------
extern "C" void kernel_launch(void* const* d_in, const int* in_sizes, int n_in,
                              void* d_out, int out_size, void* d_ws, size_t ws_size,
                              hipStream_t stream)
{
    const float* x     = (const float*)d_in[0];
    const float* Wf    = (const float*)d_in[1];
    const float* bfv   = (const float*)d_in[2];
    const float* Wg    = (const float*)d_in[3];
    const float* bg    = (const float*)d_in[4];
    const float* Wh    = (const float*)d_in[5];
    const float* bh    = (const float*)d_in[6];
    const float* Wo    = (const float*)d_in[7];
    const float* bo    = (const float*)d_in[8];
    const float* gamma = (const float*)d_in[9];
    float* out = (float*)d_out;

    // Workspace (~25.1 MB; all intermediates fit MI455X's 192 MB L2)
    char* ws = (char*)d_ws;
    size_t off = 0;
    float* F     = (float*)(ws + off); off += (size_t)16 * 16 * 1024 * 4;  // 1 MB
    float* G     = (float*)(ws + off); off += (size_t)16 * 16 * 4096 * 4;  // 4 MB
    float* V     = (float*)(ws + off); off += (size_t)16 * 64 * 1024 * 4;  // 4 MB (becomes U)
    float* rMax  = (float*)(ws + off); off += (size_t)16 * 1024 * 4;       // 64 KB
    float* rInvZ = (float*)(ws + off); off += (size_t)16 * 1024 * 4;       // 64 KB
    float* O     = (float*)(ws + off); off += (size_t)16 * 64 * 4096 * 4;  // 16 MB
    (void)ws_size; (void)in_sizes; (void)n_in; (void)out_size;

    proj_g_kernel   <<<dim3(4, 16, 16),   256, 0, stream>>>(x, Wg, bg, G);
    proj_pool_kernel<<<dim3(1, 16, 16),   256, 0, stream>>>(x, Wf, bfv, F, 16);
    proj_pool_kernel<<<dim3(1, 64, 16),   256, 0, stream>>>(x, Wh, bh, V, 64);
    row_stats_kernel<<<dim3(64, 16),      256, 0, stream>>>(F, G, rMax, rInvZ);
    scale_v_kernel  <<<dim3(4, 64, 16),   256, 0, stream>>>(V, rInvZ);
    attend_kernel   <<<dim3(32, 16),      256, 0, stream>>>(F, G, V, rMax, O);
    out_proj_kernel <<<dim3(4, 128, 16),  256, 0, stream>>>(O, Wo, bo, gamma, x, out);
}